// StructuredLinear_78958678769810
// MI455X (gfx1250) — compile-verified
//
#include <hip/hip_runtime.h>

typedef __attribute__((ext_vector_type(2))) float v2f;
typedef __attribute__((ext_vector_type(8))) float v8f;

#define IN_F 4096
#define ROWS 8192
#define BLK 64           // src/dst block size
#define NBLK 64          // number of diagonal blocks
#define ROWS_PER_WG 128  // 8 waves * 16 rows

// y[n, o] = sum_{i in block(o)} x[n,i] * W[o,i]
// Per block b: [8192 x 64] @ [64 x 64]^T -> [8192 x 64], done as
// 16x16 output tiles with 16 chained v_wmma_f32_16x16x4_f32 (K=64).
__global__ __launch_bounds__(256)
void structured_linear_wmma(const float* __restrict__ x,
                            const float* __restrict__ w,
                            float* __restrict__ y)
{
    const int cb   = blockIdx.y;        // diagonal block index (0..63)
    const int rowg = blockIdx.x;        // 128-row group (0..63)
    const int tid  = threadIdx.x;
    const int wave = tid >> 5;          // 0..7
    const int lane = tid & 31;
    const int h    = lane >> 4;         // lane half: selects K-pair
    const int l16  = lane & 15;         // M (for A) / N (for B) index

    const int row_base = rowg * ROWS_PER_WG + wave * 16;
    const int colb     = cb * BLK;

    // ---- A: x tile, 16 rows x K=64, WMMA 16x16x4-f32 layout ----
    // chunk c covers K = [4c, 4c+4): lane holds K = 4c + 2h + {0,1}
    v2f a[16];
    const float* xrow = x + (size_t)(row_base + l16) * IN_F + colb + 2 * h;
    #pragma unroll
    for (int c = 0; c < 16; ++c)
        a[c] = *(const v2f*)(xrow + 4 * c);

    // ---- 4 column tiles of the 64-wide block ----
    #pragma unroll
    for (int ct = 0; ct < 4; ++ct) {
        v8f acc = {};
        // B[k][n] = W[colb + ct*16 + n][colb + k]  (Wb transposed)
        // lane holds K = 4c + 2h + {0,1} for output col n = ct*16 + l16
        const float* wrow =
            w + (size_t)(colb + ct * 16 + l16) * IN_F + colb + 2 * h;
        #pragma unroll
        for (int c = 0; c < 16; ++c) {
            v2f b = *(const v2f*)(wrow + 4 * c);
            acc = __builtin_amdgcn_wmma_f32_16x16x4_f32(
                /*neg_a=*/false, a[c], /*neg_b=*/false, b,
                /*c_mod=*/(short)0, acc, /*reuse_a=*/false, /*reuse_b=*/false);
        }
        // ---- store 16x16 tile: VGPR v -> row (v + 8h), col ct*16 + l16 ----
        float* yo = y + (size_t)row_base * IN_F + colb + ct * 16 + l16;
        #pragma unroll
        for (int v = 0; v < 8; ++v)
            yo[(size_t)(v + 8 * h) * IN_F] = acc[v];
    }
}

extern "C" void kernel_launch(void* const* d_in, const int* in_sizes, int n_in,
                              void* d_out, int out_size, void* d_ws, size_t ws_size,
                              hipStream_t stream) {
    (void)in_sizes; (void)n_in; (void)out_size; (void)d_ws; (void)ws_size;
    const float* x = (const float*)d_in[0];   // [8192, 4096] fp32
    const float* w = (const float*)d_in[1];   // [4096, 4096] fp32
    float*       y = (float*)d_out;           // [8192, 4096] fp32

    dim3 grid(ROWS / ROWS_PER_WG, NBLK);      // (64, 64)
    structured_linear_wmma<<<grid, 256, 0, stream>>>(x, w, y);
}